// ContactGNN_9036611191377
// MI455X (gfx1250) — compile-verified
//
#include <hip/hip_runtime.h>
#include <math.h>

#define NN 100000
#define NE 1600000
#define DIN 64
#define DHID 64
#define DOUT 32

typedef float v2f __attribute__((ext_vector_type(2)));
typedef float v8f __attribute__((ext_vector_type(8)));

// ---------------------------------------------------------------------------
// 1) degree accumulation: deg[col[e]] += w[e]
// ---------------------------------------------------------------------------
__global__ void deg_kernel(const int* __restrict__ col,
                           const float* __restrict__ w,
                           float* __restrict__ deg, int E) {
    int e = blockIdx.x * blockDim.x + threadIdx.x;
    if (e < E) atomicAdd(&deg[col[e]], w[e]);
}

// 2) dinv = deg > 0 ? rsqrt(deg) : 0   (in place)
__global__ void dinv_kernel(float* __restrict__ deg, int n) {
    int i = blockIdx.x * blockDim.x + threadIdx.x;
    if (i < n) {
        float d = deg[i];
        deg[i] = (d > 0.0f) ? rsqrtf(d) : 0.0f;
    }
}

// 3) norm[e] = dinv[row[e]] * w[e] * dinv[col[e]]
__global__ void norm_kernel(const int* __restrict__ row,
                            const int* __restrict__ col,
                            const float* __restrict__ w,
                            const float* __restrict__ dinv,
                            float* __restrict__ norm, int E) {
    int e = blockIdx.x * blockDim.x + threadIdx.x;
    if (e < E) norm[e] = dinv[row[e]] * w[e] * dinv[col[e]];
}

// ---------------------------------------------------------------------------
// 4) Dense GEMM via V_WMMA_F32_16X16X4_F32.
//    H[n x F] = act(A[n x K] (+bias, relu)) @ W[K x F]
//    One wave computes one 16x16 tile of H; block of (F/16) waves covers a
//    16-row stripe across all F output columns. K loop steps by 4.
//    WMMA operand layouts per CDNA5 ISA 7.12.2 (wave32):
//      A (16x4 f32, v2f): lane l -> M = l&15 ; half = l>>4 ;
//                         a.x = A[M][k+2*half], a.y = A[M][k+2*half+1]
//      B (4x16 f32, v2f): lane l -> N = l&15 ; b.x = W[k+2*half][N], b.y = W[k+2*half+1][N]
//      C/D (16x16 f32, v8f): c[j] = H[8*half + j][l&15]
// ---------------------------------------------------------------------------
template <int K, int F, bool BIAS_RELU>
__global__ void gemm_wmma_f32(const float* __restrict__ A,
                              const float* __restrict__ W,
                              const float* __restrict__ bias,
                              float* __restrict__ H) {
    const int lane = threadIdx.x & 31;
    const int wave = threadIdx.x >> 5;        // column tile index
    const int half = lane >> 4;               // 0: K pair 0/1, 1: K pair 2/3
    const int ml   = lane & 15;               // row (A) / col (B) within tile
    const int m0   = blockIdx.x * 16;         // row tile base
    const int nb   = wave * 16;               // col tile base

    const float* __restrict__ arow = A + (size_t)(m0 + ml) * K;

    v8f c = {};
#pragma unroll
    for (int k = 0; k < K; k += 4) {
        const int kb = k + half * 2;
        float a0 = arow[kb];
        float a1 = arow[kb + 1];
        if (BIAS_RELU) {
            a0 = fmaxf(a0 + bias[kb], 0.0f);
            a1 = fmaxf(a1 + bias[kb + 1], 0.0f);
        }
        v2f a, b;
        a.x = a0;
        a.y = a1;
        b.x = W[(size_t)kb * F + nb + ml];
        b.y = W[(size_t)(kb + 1) * F + nb + ml];
        c = __builtin_amdgcn_wmma_f32_16x16x4_f32(
                /*neg_a=*/false, a, /*neg_b=*/false, b,
                /*c_mod=*/(short)0, c, /*reuse_a=*/false, /*reuse_b=*/false);
    }

#pragma unroll
    for (int j = 0; j < 8; ++j) {
        const int m = m0 + half * 8 + j;
        H[(size_t)m * F + nb + ml] = c[j];
    }
}

// ---------------------------------------------------------------------------
// 5) Edge scatter: out[col[e]][:] += norm[e] * h[row[e]][:]
//    One wave per edge. F=64: 2 dims/lane via float2 row read (coalesced 256B),
//    F=32: 1 dim/lane. Atomic f32 adds resolve in L2 (agg arrays ~25MB << 192MB).
// ---------------------------------------------------------------------------
template <int F>
__global__ void scatter_kernel(const int* __restrict__ row,
                               const int* __restrict__ col,
                               const float* __restrict__ norm,
                               const float* __restrict__ h,
                               float* __restrict__ out, int E) {
    const int wavesPerBlock = blockDim.x >> 5;
    const int e = blockIdx.x * wavesPerBlock + (threadIdx.x >> 5);
    if (e >= E) return;
    const int lane = threadIdx.x & 31;
    const int r = row[e];
    const int cc = col[e];
    const float nv = norm[e];
    if (F == 64) {
        const float2* __restrict__ src =
            (const float2*)(h + (size_t)r * 64);
        float2 v = src[lane];
        float* dst = out + (size_t)cc * 64 + 2 * lane;
        atomicAdd(dst + 0, nv * v.x);
        atomicAdd(dst + 1, nv * v.y);
    } else {
        float v = h[(size_t)r * 32 + lane];
        atomicAdd(out + (size_t)cc * 32 + lane, nv * v);
    }
}

// 6) final: out = sigmoid(out + b2)   (in place over N x 32)
__global__ void sigmoid_bias_kernel(float* __restrict__ out,
                                    const float* __restrict__ bias, int total) {
    int i = blockIdx.x * blockDim.x + threadIdx.x;
    if (i < total) {
        float v = out[i] + bias[i & (DOUT - 1)];
        out[i] = 1.0f / (1.0f + __expf(-v));
    }
}

// ---------------------------------------------------------------------------
extern "C" void kernel_launch(void* const* d_in, const int* in_sizes, int n_in,
                              void* d_out, int out_size, void* d_ws, size_t ws_size,
                              hipStream_t stream) {
    const float* x    = (const float*)d_in[0];   // [N, 64]
    const int*   ei   = (const int*)d_in[1];     // [2, E] : row = ei[0:E], col = ei[E:2E]
    const float* ew   = (const float*)d_in[2];   // [E]
    const float* W1   = (const float*)d_in[3];   // [64, 64]
    const float* b1   = (const float*)d_in[4];   // [64]
    const float* W2   = (const float*)d_in[5];   // [64, 32]
    const float* b2   = (const float*)d_in[6];   // [32]
    float*       out  = (float*)d_out;           // [N, 32]

    const int* row = ei;
    const int* col = ei + NE;

    // workspace carve-up (all float32)
    float* dinv = (float*)d_ws;                  // N          (deg -> dinv in place)
    float* norm = dinv + NN;                     // E
    float* h1   = norm + NE;                     // N*64
    float* agg1 = h1 + (size_t)NN * DHID;        // N*64
    float* h2   = agg1 + (size_t)NN * DHID;      // N*32

    // zero the accumulators every call (harness does not re-poison)
    hipMemsetAsync(dinv, 0, (size_t)NN * sizeof(float), stream);
    hipMemsetAsync(agg1, 0, (size_t)NN * DHID * sizeof(float), stream);
    hipMemsetAsync(out,  0, (size_t)NN * DOUT * sizeof(float), stream);

    // 1) degree
    deg_kernel<<<(NE + 255) / 256, 256, 0, stream>>>(col, ew, dinv, NE);
    // 2) dinv
    dinv_kernel<<<(NN + 255) / 256, 256, 0, stream>>>(dinv, NN);
    // 3) norm
    norm_kernel<<<(NE + 255) / 256, 256, 0, stream>>>(row, col, ew, dinv, norm, NE);

    // 4) h1 = x @ W1  (N=100000 = 6250*16 exactly)
    gemm_wmma_f32<DIN, DHID, false>
        <<<NN / 16, (DHID / 16) * 32, 0, stream>>>(x, W1, nullptr, h1);

    // 5) agg1[col] += norm * h1[row]   (wave per edge, 8 waves / block)
    scatter_kernel<DHID>
        <<<(NE + 7) / 8, 256, 0, stream>>>(row, col, norm, h1, agg1, NE);

    // 6) h2 = relu(agg1 + b1) @ W2   (bias+relu fused into A-operand load)
    gemm_wmma_f32<DHID, DOUT, true>
        <<<NN / 16, (DOUT / 16) * 32, 0, stream>>>(agg1, W2, b1, h2);

    // 7) out[col] += norm * h2[row]
    scatter_kernel<DOUT>
        <<<(NE + 7) / 8, 256, 0, stream>>>(row, col, norm, h2, out, NE);

    // 8) out = sigmoid(out + b2)
    sigmoid_bias_kernel<<<(NN * DOUT + 255) / 256, 256, 0, stream>>>(out, b2, NN * DOUT);
}